// MRTE_31662498906764
// MI455X (gfx1250) — compile-verified
//
#include <hip/hip_runtime.h>
#include <stdint.h>

// Problem constants (fixed by the reference file).
#define BB 32
#define LL 1024
#define DD 512            // floats per row (2048 bytes)
#define TT 7168           // mel_max_length
#define WAVES_PER_BLOCK 8 // 256 threads = 8 wave32

// ---------------------------------------------------------------------------
// Kernel A: per-batch inclusive scan of durations + scatter of covering token
// index into idx[b, t] (int32). Frames >= total get -1 (=> zero output row).
// One block of 1024 threads per batch; scan done in LDS (Hillis-Steele).
// ---------------------------------------------------------------------------
__global__ __launch_bounds__(LL) void lr_build_idx(const int* __restrict__ dur,
                                                   int* __restrict__ idx) {
  __shared__ int cum[LL];
  const int b = blockIdx.x;
  const int j = threadIdx.x;

  cum[j] = dur[b * LL + j];
  __syncthreads();

  // Inclusive scan over 1024 elements, 10 steps.
  #pragma unroll
  for (int off = 1; off < LL; off <<= 1) {
    int add = (j >= off) ? cum[j - off] : 0;
    __syncthreads();
    cum[j] += add;
    __syncthreads();
  }

  const int start = (j == 0) ? 0 : cum[j - 1];
  const int end   = cum[j];
  int* rowIdx = idx + (size_t)b * TT;

  // Token j covers frames [start, end). Disjoint ranges across threads.
  for (int t = start; t < end; ++t) {
    rowIdx[t] = j;
  }

  // Tail fill: frames >= total are invalid (-1). Disjoint from scatter above.
  const int total = cum[LL - 1];
  for (int t = total + j; t < TT; t += LL) {
    rowIdx[t] = -1;
  }
}

// ---------------------------------------------------------------------------
// Kernel B: gather-copy. One wave32 per output row of 512 floats (2 KB).
// Valid rows are moved via the CDNA5 async DMA path:
//   global_load_async_to_lds_b128  (HBM/L2 -> LDS slice, ASYNCcnt)
//   s_wait_asynccnt 0
//   global_store_async_from_lds_b128 (LDS slice -> HBM, ASYNCcnt)
// Each lane handles 4 x 16B chunks => 4 async issues each way per lane.
// Invalid rows (j < 0) are zero-filled with plain B128 stores.
// j is uniform per wave (one row per wave) so EXEC stays all-ones around the
// async ops, as required. Kernel-end S_ENDPGM implicitly waits all counters.
// ---------------------------------------------------------------------------
__global__ __launch_bounds__(WAVES_PER_BLOCK * 32) void lr_gather_copy(
    const float* __restrict__ x, const int* __restrict__ idx,
    float* __restrict__ out) {
  __shared__ float lds[WAVES_PER_BLOCK * DD];  // 16 KB: 2 KB slice per wave

  const int wave = threadIdx.x >> 5;
  const int lane = threadIdx.x & 31;
  const long row = (long)blockIdx.x * WAVES_PER_BLOCK + wave;  // < BB*TT exactly

  const int b = (int)(row / TT);
  const int j = idx[row];

  float* __restrict__ dst = out + row * DD;

  if (j < 0) {
    // Invalid frame: write a zero row (4 x B128 per lane, coalesced).
    const float4 z = make_float4(0.f, 0.f, 0.f, 0.f);
    float4* d4 = (float4*)dst;
    #pragma unroll
    for (int k = 0; k < 4; ++k) {
      d4[lane + 32 * k] = z;
    }
    return;
  }

  const float* __restrict__ src = x + ((size_t)b * LL + (size_t)j) * DD;

  // LDS byte offset of this wave's slice: low 32 bits of the generic address
  // (ISA 10.2: flat->LDS address is addr[31:0], relative to wave LDS base).
  const unsigned lbase = (unsigned)(uintptr_t)(&lds[wave * DD]);

  // Stage row into LDS via async DMA (4 x 512B per wave-issue).
  #pragma unroll
  for (int k = 0; k < 4; ++k) {
    unsigned laddr = lbase + (unsigned)(lane * 16 + k * 512);
    unsigned long long gaddr =
        (unsigned long long)(uintptr_t)(src + lane * 4 + k * 128);
    asm volatile("global_load_async_to_lds_b128 %0, %1, off"
                 :: "v"(laddr), "v"(gaddr)
                 : "memory");
  }

  // Wait for the LDS writes of the async loads before reading LDS back out.
  asm volatile("s_wait_asynccnt 0" ::: "memory");

  // Stream row out of LDS to the destination via async DMA.
  #pragma unroll
  for (int k = 0; k < 4; ++k) {
    unsigned laddr = lbase + (unsigned)(lane * 16 + k * 512);
    unsigned long long gaddr =
        (unsigned long long)(uintptr_t)(dst + lane * 4 + k * 128);
    asm volatile("global_store_async_from_lds_b128 %0, %1, off"
                 :: "v"(gaddr), "v"(laddr)
                 : "memory");
  }
  // Implicit S_WAIT_IDLE at S_ENDPGM covers outstanding ASYNCcnt stores.
}

extern "C" void kernel_launch(void* const* d_in, const int* in_sizes, int n_in,
                              void* d_out, int out_size, void* d_ws, size_t ws_size,
                              hipStream_t stream) {
  const float* x   = (const float*)d_in[0];       // [B, L, D] f32
  const int*   dur = (const int*)d_in[1];         // [B, L] i32
  // d_in[2] = mel_max_length (device scalar) -- value is TT, known statically.

  float* out = (float*)d_out;                     // [B, T, D] f32
  int*   idx = (int*)d_ws;                        // [B, T] i32 scratch (896 KB)

  // Pass 1: build covering-token index map.
  lr_build_idx<<<BB, LL, 0, stream>>>(dur, idx);

  // Pass 2: expand rows. One wave per row; 8 waves per block.
  const int rows   = BB * TT;                     // 229376
  const int blocks = rows / WAVES_PER_BLOCK;      // 28672 (exact)
  lr_gather_copy<<<blocks, WAVES_PER_BLOCK * 32, 0, stream>>>(x, idx, out);
}